// TopkSparseAutoEncoder2Child_v2_51642686767578
// MI455X (gfx1250) — compile-verified
//
#include <hip/hip_runtime.h>
#include <hip/hip_bf16.h>
#include <math.h>

#define B_SZ 2048
#define D_SZ 512
#define F_SZ 16384
#define AUX_THRESH 0.7f
#define EPS_F 1e-8f

typedef float v2f __attribute__((ext_vector_type(2)));
typedef float v4f __attribute__((ext_vector_type(4)));
typedef float v8f __attribute__((ext_vector_type(8)));
typedef unsigned long long u64;

#define XS_STRIDE 516            // 512 + 4 pad: lanes hit banks 4*l -> conflict-free b64
#define SPLIT_F 4
#define FSLICE (F_SZ / SPLIT_F)  // 4096 features per WG
#define WAVES 8
#define TILES_PER_WAVE (FSLICE / 16 / WAVES) // 32

// ws layout (4-byte words):
// [0,F) pp  [F,2F) p1  [2F,3F) p2  [3F,4F) s11  [4F,5F) s22
// [5F,6F) parent flags  [6F,7F) c1 flags  [7F,8F) c2 flags  (u32)
// [8F] aux_sum  [8F+1] active_cnt
// [8F+2 ..) cand keys: u64[B*12]  (byte offset (8F+2)*4 is 8B aligned)

// ---- sortable packed key: (value desc, index asc) == u64 descending ----
__device__ __forceinline__ u64 pack_key(float v, int i) {
  unsigned int b = __float_as_uint(v);
  // monotone float->uint mapping: pos -> b|0x80000000-ish, neg -> ~b
  unsigned int ob = b ^ ((unsigned int)((int)b >> 31) | 0x80000000u);
  return ((u64)ob << 32) | (unsigned int)(~i);
}
__device__ __forceinline__ float key_val(u64 k) {
  unsigned int ob = (unsigned int)(k >> 32);
  unsigned int b = (ob & 0x80000000u) ? (ob ^ 0x80000000u) : ~ob;
  return __uint_as_float(b);
}
__device__ __forceinline__ int key_idx(u64 k) {
  return (int)~(unsigned int)k;
}
// branch-free top-3: 3 u64 compares + cndmask chains
__device__ __forceinline__ void top3_insert(u64 k, u64& k0, u64& k1, u64& k2) {
  bool b0 = k > k0;
  bool b1 = k > k1;
  bool b2 = k > k2;
  u64 n2 = b1 ? k1 : (b2 ? k : k2);
  u64 n1 = b0 ? k0 : (b1 ? k : k1);
  u64 n0 = b0 ? k  : k0;
  k0 = n0; k1 = n1; k2 = n2;
}

__global__ __launch_bounds__(256) void k_zero(float* p, int n) {
  int t = blockIdx.x * 256 + threadIdx.x;
  if (t < n) p[t] = 0.0f;
}

// per-feature Gram terms; thread = feature -> coalesced across F
__global__ __launch_bounds__(256) void k_gram(
    const float* __restrict__ Wd, const float* __restrict__ Wd1,
    const float* __restrict__ Wd2, float* __restrict__ ws) {
  int f = blockIdx.x * 256 + threadIdx.x;
  float pp = 0.f, p1 = 0.f, p2 = 0.f, s11 = 0.f, s22 = 0.f;
#pragma unroll 4
  for (int d = 0; d < D_SZ; ++d) {
    size_t off = (size_t)d * F_SZ + f;
    float a = Wd[off], b = Wd1[off], c = Wd2[off];
    pp  = fmaf(a, a, pp);  p1  = fmaf(a, b, p1);  p2 = fmaf(a, c, p2);
    s11 = fmaf(b, b, s11); s22 = fmaf(c, c, s22);
  }
  ws[f] = pp; ws[F_SZ + f] = p1; ws[2 * F_SZ + f] = p2;
  ws[3 * F_SZ + f] = s11; ws[4 * F_SZ + f] = s22;
}

// WMMA fp32 encoder + streaming per-lane top-3.
// Grid: (B/16) row-tiles x SPLIT_F feature slices. Block: 256 (8 waves).
// C tile is pre^T: M = feature (VGPR axis), N = batch (lane axis), so each
// lane owns one batch row's values -> top-3 update is pure per-lane VALU.
__global__ __launch_bounds__(256) void k_encode_top3(
    const float* __restrict__ x, const float* __restrict__ We,
    const float* __restrict__ be, u64* __restrict__ cand_k) {
  __shared__ float xs[16 * XS_STRIDE];
  __shared__ u64 ck[256 * 3];

  const int tid = threadIdx.x;
  const int row_tile = blockIdx.x >> 2;
  const int split = blockIdx.x & (SPLIT_F - 1);
  const int row0 = row_tile * 16;

  // stage x tile (16 x 512) with padded stride
  for (int t = tid; t < 16 * D_SZ; t += 256) {
    int r = t >> 9, d = t & (D_SZ - 1);
    xs[r * XS_STRIDE + d] = x[(size_t)(row0 + r) * D_SZ + d];
  }
  __syncthreads();

  const int wave = tid >> 5;
  const int lane = tid & 31;
  const int half = lane >> 4;        // f32 WMMA A: lanes 0-15 K={0,1}, 16-31 K={2,3}
  const int mrow = lane & 15;
  const int koff = half << 1;
  const float* xrow = xs + mrow * XS_STRIDE + koff;   // B frag: lane = batch col

  u64 k0 = 0ull, k1 = 0ull, k2 = 0ull;   // 0 = -inf sentinel (any real key wins)

  for (int t = 0; t < TILES_PER_WAVE; ++t) {
    const int f0 = (split * (FSLICE / 16) + wave + t * WAVES) << 4;
    const float* arow = We + (size_t)(f0 + mrow) * D_SZ + koff; // A frag: lane = feature row
    v8f acc = {};
#pragma unroll 8
    for (int k = 0; k < D_SZ; k += 4) {
      v2f af = *(const v2f*)(arow + k);   // global_load_b64 of We
      v2f bf = *(const v2f*)(xrow + k);   // ds_load_b64 of staged x
      acc = __builtin_amdgcn_wmma_f32_16x16x4_f32(
          false, af, false, bf, (short)0, acc, false, false);
    }
    const int fbase = f0 + (half << 3);   // C: lanes>=16 hold M = v+8
    v4f blo = *(const v4f*)(be + fbase);       // global_load_b128 bias blocks
    v4f bhi = *(const v4f*)(be + fbase + 4);
#pragma unroll
    for (int v = 0; v < 8; ++v) {
      float bias = (v < 4) ? blo[v & 3] : bhi[v & 3];
      float pv = acc[v] + bias;
      top3_insert(pack_key(pv, fbase + v), k0, k1, k2);
    }
  }

  int cb = tid * 3;
  ck[cb] = k0; ck[cb + 1] = k1; ck[cb + 2] = k2;
  __syncthreads();

  if (tid < 16) {  // merge 8 waves x 2 lane-halves for batch row `tid`
    u64 m0 = 0ull, m1 = 0ull, m2 = 0ull;
    for (int w = 0; w < WAVES; ++w)
      for (int h = 0; h < 2; ++h) {
        int base = (w * 32 + h * 16 + tid) * 3;
#pragma unroll
        for (int c = 0; c < 3; ++c)
          top3_insert(ck[base + c], m0, m1, m2);
      }
    int ob = (row0 + tid) * (SPLIT_F * 3) + split * 3;
    cand_k[ob] = m0; cand_k[ob + 1] = m1; cand_k[ob + 2] = m2;
  }
}

// one wave per batch row: merge candidates, child dots, routing, flags,
// aux-cos terms (via precomputed Gram), sparse recon gather.
__global__ __launch_bounds__(256) void k_rowfinal(
    const float* __restrict__ x,
    const float* __restrict__ We1, const float* __restrict__ be1,
    const float* __restrict__ We2, const float* __restrict__ be2,
    const float* __restrict__ Wd,  const float* __restrict__ bd,
    const float* __restrict__ Wd1, const float* __restrict__ bd1,
    const float* __restrict__ Wd2, const float* __restrict__ bd2,
    const u64* __restrict__ cand_k,
    float* __restrict__ ws, float* __restrict__ out) {
  const int tid = threadIdx.x;
  const int wave = tid >> 5;
  const int lane = tid & 31;
  const int b = blockIdx.x * 8 + wave;

  // all lanes redundantly merge the 12 candidates (broadcast loads)
  u64 m0 = 0ull, m1k = 0ull, m2k = 0ull;
  const int cb = b * (SPLIT_F * 3);
#pragma unroll
  for (int c = 0; c < SPLIT_F * 3; ++c)
    top3_insert(cand_k[cb + c], m0, m1k, m2k);
  float vk[3] = {key_val(m0), key_val(m1k), key_val(m2k)};
  int   ik[3] = {key_idx(m0), key_idx(m1k), key_idx(m2k)};

  // child pre-activations at the 3 selected features
  float pre1[3], pre2[3];
  const float* xr = x + (size_t)b * D_SZ;
#pragma unroll
  for (int k = 0; k < 3; ++k) {
    const float* w1 = We1 + (size_t)ik[k] * D_SZ;
    const float* w2 = We2 + (size_t)ik[k] * D_SZ;
    float a1 = 0.f, a2 = 0.f;
    for (int d = lane; d < D_SZ; d += 32) {
      float xv = xr[d];
      a1 = fmaf(xv, w1[d], a1);
      a2 = fmaf(xv, w2[d], a2);
    }
#pragma unroll
    for (int off = 16; off > 0; off >>= 1) {   // butterfly: all lanes get sum
      a1 += __shfl_xor(a1, off, 32);
      a2 += __shfl_xor(a2, off, 32);
    }
    pre1[k] = a1 + be1[ik[k]];
    pre2[k] = a2 + be2[ik[k]];
  }

  float fa1[3], fa2[3], csel[3];
  bool win[3];
#pragma unroll
  for (int k = 0; k < 3; ++k) {
    bool nz = vk[k] != 0.f;
    float m1 = nz ? pre1[k] : 0.f;
    float m2 = nz ? pre2[k] : 0.f;
    win[k] = m1 > m2;
    fa1[k] = win[k] ? m1 : 0.f;
    fa2[k] = win[k] ? 0.f : m2;
    csel[k] = win[k] ? fa1[k] : fa2[k];
  }

  unsigned* pflag  = (unsigned*)(ws + 5 * F_SZ);
  unsigned* c1flag = (unsigned*)(ws + 6 * F_SZ);
  unsigned* c2flag = (unsigned*)(ws + 7 * F_SZ);
  float* aux_acc = ws + 8 * F_SZ;
  if (lane < 3) {
    int k = lane, i = ik[k];
    pflag[i] = 1u;                       // .at[idx].set(1.0) is unconditional
    if (fa1[k] != 0.f) c1flag[i] = 1u;
    if (fa2[k] != 0.f) c2flag[i] = 1u;
    float a = vk[k];
    if (a > 0.f) {
      float ppv = ws[i];
      float pcv = win[k] ? ws[F_SZ + i]     : ws[2 * F_SZ + i];
      float ccv = win[k] ? ws[3 * F_SZ + i] : ws[4 * F_SZ + i];
      float c = csel[k];
      float dot = a * (a * ppv + c * pcv);
      float nx  = fabsf(a) * sqrtf(ppv);
      float ny2 = a * a * ppv + 2.f * a * c * pcv + c * c * ccv;
      float ny  = sqrtf(fmaxf(ny2, 0.f));
      float cosv = dot / (fmaxf(nx, EPS_F) * fmaxf(ny, EPS_F));
      float sim = fmaxf(0.f, AUX_THRESH - cosv);
      atomicAdd(aux_acc, sim);
      atomicAdd(aux_acc + 1, 1.0f);
    }
  }

  // sparse reconstruction: recon[b,d] = bd+bd1+bd2 + sum_k contributions
  for (int d = lane; d < D_SZ; d += 32) {
    float acc = bd[d] + bd1[d] + bd2[d];
#pragma unroll
    for (int k = 0; k < 3; ++k) {
      size_t off = (size_t)d * F_SZ + ik[k];
      if (vk[k]  != 0.f) acc = fmaf(vk[k],  Wd[off],  acc);
      if (fa1[k] != 0.f) acc = fmaf(fa1[k], Wd1[off], acc);
      if (fa2[k] != 0.f) acc = fmaf(fa2[k], Wd2[off], acc);
    }
    out[(size_t)b * D_SZ + d] = acc;
  }
}

__global__ __launch_bounds__(256) void k_finalize(const float* __restrict__ ws,
                                                  float* __restrict__ out) {
  __shared__ int s0[256], s1[256], s2[256];
  const unsigned* pf = (const unsigned*)(ws + 5 * F_SZ);
  const unsigned* f1 = (const unsigned*)(ws + 6 * F_SZ);
  const unsigned* f2 = (const unsigned*)(ws + 7 * F_SZ);
  int t = threadIdx.x;
  int c0 = 0, c1 = 0, c2 = 0;
  for (int f = t; f < F_SZ; f += 256) {
    c0 += pf[f] ? 1 : 0; c1 += f1[f] ? 1 : 0; c2 += f2[f] ? 1 : 0;
  }
  s0[t] = c0; s1[t] = c1; s2[t] = c2;
  __syncthreads();
  for (int s = 128; s > 0; s >>= 1) {
    if (t < s) { s0[t] += s0[t + s]; s1[t] += s1[t + s]; s2[t] += s2[t + s]; }
    __syncthreads();
  }
  if (t == 0) {
    int* io = (int*)(out + (size_t)B_SZ * D_SZ);
    io[0] = s0[0]; io[1] = s1[0]; io[2] = s2[0];
    out[(size_t)B_SZ * D_SZ + 3] = ws[8 * F_SZ] / (ws[8 * F_SZ + 1] + 1e-8f);
  }
}

extern "C" void kernel_launch(void* const* d_in, const int* in_sizes, int n_in,
                              void* d_out, int out_size, void* d_ws, size_t ws_size,
                              hipStream_t stream) {
  (void)in_sizes; (void)n_in; (void)out_size; (void)ws_size;
  const float* x   = (const float*)d_in[0];
  const float* We  = (const float*)d_in[1];
  const float* be  = (const float*)d_in[2];
  const float* Wd  = (const float*)d_in[3];
  const float* bd  = (const float*)d_in[4];
  const float* We1 = (const float*)d_in[5];
  const float* be1 = (const float*)d_in[6];
  const float* Wd1 = (const float*)d_in[7];
  const float* bd1 = (const float*)d_in[8];
  const float* We2 = (const float*)d_in[9];
  const float* be2 = (const float*)d_in[10];
  const float* Wd2 = (const float*)d_in[11];
  const float* bd2 = (const float*)d_in[12];
  float* out = (float*)d_out;
  float* ws  = (float*)d_ws;
  u64* cand_k = (u64*)(ws + 8 * F_SZ + 2);   // byte offset divisible by 8

  int nzero = 3 * F_SZ + 2;  // flags + accumulators
  hipLaunchKernelGGL(k_zero, dim3((nzero + 255) / 256), dim3(256), 0, stream,
                     ws + 5 * F_SZ, nzero);
  hipLaunchKernelGGL(k_gram, dim3(F_SZ / 256), dim3(256), 0, stream,
                     Wd, Wd1, Wd2, ws);
  hipLaunchKernelGGL(k_encode_top3, dim3((B_SZ / 16) * SPLIT_F), dim3(256), 0,
                     stream, x, We, be, cand_k);
  hipLaunchKernelGGL(k_rowfinal, dim3(B_SZ / 8), dim3(256), 0, stream,
                     x, We1, be1, We2, be2, Wd, bd, Wd1, bd1, Wd2, bd2,
                     cand_k, ws, out);
  hipLaunchKernelGGL(k_finalize, dim3(1), dim3(256), 0, stream, ws, out);
}